// SMGCN_73272142069947
// MI455X (gfx1250) — compile-verified
//
#include <hip/hip_runtime.h>
#include <hip/hip_bf16.h>
#include <math.h>

typedef __attribute__((ext_vector_type(2))) float v2f;
typedef __attribute__((ext_vector_type(8))) float v8f;

#define D_DIM 256
#define KC 32
#define LDA_S (KC + 2)      // 34 floats: even -> v2f loads stay 8B aligned
#define LDBT  (KC + 2)      // Bst[n][k] row stride (34)

// ---------------------------------------------------------------------------
// zero fill (float4 granularity; all sizes here are multiples of 4)
// ---------------------------------------------------------------------------
__global__ __launch_bounds__(256)
void zero_f32(float4* __restrict__ p, long n4) {
    long i = (long)blockIdx.x * blockDim.x + threadIdx.x;
    if (i < n4) p[i] = make_float4(0.f, 0.f, 0.f, 0.f);
}

// ---------------------------------------------------------------------------
// SPMM: Y[rows[e]] += vals[e] * X[cols[e]]   (one wave32 per nnz)
// X row c comes from Xa if c < split else Xb[c-split]  (pre0 concat handling)
// ---------------------------------------------------------------------------
__global__ __launch_bounds__(256)
void spmm_atomic(const int* __restrict__ rows, const int* __restrict__ cols,
                 const float* __restrict__ vals,
                 const float* __restrict__ Xa, const float* __restrict__ Xb,
                 int split, float* __restrict__ Y, int nnz) {
    int w    = blockIdx.x * 8 + (threadIdx.x >> 5);
    int lane = threadIdx.x & 31;
    if (w >= nnz) return;
    int   r = rows[w];
    int   c = cols[w];
    float v = vals[w];
    const float* X = (c < split) ? (Xa + (size_t)c * D_DIM)
                                 : (Xb + (size_t)(c - split) * D_DIM);
    float* y = Y + (size_t)r * D_DIM;
#pragma unroll
    for (int i = 0; i < 2; ++i) {
        int    b = (lane + 32 * i) * 4;
        float4 x = *(const float4*)(X + b);
        atomicAdd(y + b + 0, x.x * v);
        atomicAdd(y + b + 1, x.y * v);
        atomicAdd(y + b + 2, x.z * v);
        atomicAdd(y + b + 3, x.w * v);
    }
}

// ---------------------------------------------------------------------------
// OUT[M x 256] (store or +=) = tanh(A0 @ B0 [+ A1 @ B1] [+ bias])
// A*: M x 256 row-major.  B*: 256 x 256 row-major.  bias: 256 (or null).
// Block: 256 threads / 8 waves; tile 64(M) x 128(N); per wave 16 x 64 via four
// V_WMMA_F32_16X16X4_F32 accumulators. B tile staged TRANSPOSED in LDS so each
// B fragment is one ds_load_b64; A fragment is one ds_load_b64 reused 4x.
// ---------------------------------------------------------------------------
__global__ __launch_bounds__(256)
void gemm_tanh_wmma(const float* __restrict__ A0, const float* __restrict__ A1,
                    const float* __restrict__ B0, const float* __restrict__ B1,
                    const float* __restrict__ bias, float* __restrict__ OUT,
                    int M, int accumulate) {
    __shared__ float As[64 * LDA_S];     // As[m][k]
    __shared__ float Bst[128 * LDBT];    // Bst[n][k]  (transposed B tile)

    const int t    = threadIdx.x;
    const int wave = t >> 5;
    const int lane = t & 31;
    const int wm   = wave & 3;        // 4 M subtiles of 16 rows
    const int wn   = wave >> 2;       // 2 N groups of 64 cols
    const int rowBase = blockIdx.x * 64;
    const int colBase = blockIdx.y * 128;

    const int m16 = lane & 15;
    const int kh  = lane >> 4;        // selects K pair {0,1} vs {2,3}

    v8f acc[4];
#pragma unroll
    for (int j = 0; j < 4; ++j) acc[j] = (v8f)0.f;

    for (int src = 0; src < 2; ++src) {
        const float* A = src ? A1 : A0;
        const float* B = src ? B1 : B0;
        if (A == nullptr) break;

        for (int k0 = 0; k0 < D_DIM; k0 += KC) {
            // stage A tile: 64 rows x KC, row-major
            {
                int r  = t >> 3;            // 0..31
                int c4 = (t & 7) * 4;       // 0..28
#pragma unroll
                for (int i = 0; i < 2; ++i) {
                    int    row = rowBase + r + 32 * i;
                    float4 v   = make_float4(0.f, 0.f, 0.f, 0.f);
                    if (row < M)
                        v = *(const float4*)(A + (size_t)row * D_DIM + k0 + c4);
                    float* dst = &As[(r + 32 * i) * LDA_S + c4];
                    dst[0] = v.x; dst[1] = v.y; dst[2] = v.z; dst[3] = v.w;
                }
            }
            // stage B tile transposed: global KC x 128 -> Bst[n][k]
            {
#pragma unroll
                for (int i = 0; i < 4; ++i) {
                    int idx = t + 256 * i;      // 0..1023
                    int k   = idx >> 5;         // 0..31
                    int n4  = (idx & 31) * 4;   // 0..124
                    float4 v = *(const float4*)(B + (size_t)(k0 + k) * D_DIM +
                                                colBase + n4);
                    Bst[(n4 + 0) * LDBT + k] = v.x;
                    Bst[(n4 + 1) * LDBT + k] = v.y;
                    Bst[(n4 + 2) * LDBT + k] = v.z;
                    Bst[(n4 + 3) * LDBT + k] = v.w;
                }
            }
            __syncthreads();

#pragma unroll
            for (int kk = 0; kk < KC; kk += 4) {
                // A frag: lane holds A[m][kk+2*kh], A[m][kk+2*kh+1]
                v2f af = *(const v2f*)&As[(wm * 16 + m16) * LDA_S + kk + 2 * kh];
#pragma unroll
                for (int j = 0; j < 4; ++j) {
                    int n  = wn * 64 + j * 16 + m16;
                    v2f bf = *(const v2f*)&Bst[n * LDBT + kk + 2 * kh];
                    acc[j] = __builtin_amdgcn_wmma_f32_16x16x4_f32(
                        false, af, false, bf, (short)0, acc[j], false, false);
                }
            }
            __syncthreads();
        }
    }

    // epilogue: lane holds D[r + 8*kh][m16] in acc[j][r]
#pragma unroll
    for (int j = 0; j < 4; ++j) {
        int   c  = colBase + wn * 64 + j * 16 + m16;
        float bv = bias ? bias[c] : 0.f;
#pragma unroll
        for (int r = 0; r < 8; ++r) {
            int row = rowBase + wm * 16 + kh * 8 + r;
            if (row < M) {
                float  v = tanhf(acc[j][r] + bv);
                float* o = OUT + (size_t)row * D_DIM + c;
                if (accumulate) *o += v;
                else            *o  = v;
            }
        }
    }
}

// ---------------------------------------------------------------------------
// row-wise L2 normalize (wave32 per row of 256 floats)
// ---------------------------------------------------------------------------
__global__ __launch_bounds__(256)
void l2norm_rows(const float* __restrict__ H, float* __restrict__ out, int M) {
    int row  = blockIdx.x * 8 + (threadIdx.x >> 5);
    int lane = threadIdx.x & 31;
    if (row >= M) return;
    const float4* h = (const float4*)(H + (size_t)row * D_DIM);
    float4 a = h[lane];
    float4 b = h[lane + 32];
    float  s = a.x * a.x + a.y * a.y + a.z * a.z + a.w * a.w +
               b.x * b.x + b.y * b.y + b.z * b.z + b.w * b.w;
#pragma unroll
    for (int off = 16; off > 0; off >>= 1) s += __shfl_xor(s, off, 32);
    float inv = 1.f / fmaxf(sqrtf(s), 1e-12f);
    a.x *= inv; a.y *= inv; a.z *= inv; a.w *= inv;
    b.x *= inv; b.y *= inv; b.z *= inv; b.w *= inv;
    float4* o = (float4*)(out + (size_t)row * D_DIM);
    o[lane]      = a;
    o[lane + 32] = b;
}

// ---------------------------------------------------------------------------
extern "C" void kernel_launch(void* const* d_in, const int* in_sizes, int n_in,
                              void* d_out, int out_size, void* d_ws, size_t ws_size,
                              hipStream_t stream) {
    const int* adj_rows   = (const int*)d_in[0];
    const int* adj_cols   = (const int*)d_in[1];
    const float* adj_vals = (const float*)d_in[2];
    const int* up_rows    = (const int*)d_in[3];
    const int* up_cols    = (const int*)d_in[4];
    const float* up_vals  = (const float*)d_in[5];
    const int* ip_rows    = (const int*)d_in[6];
    const int* ip_cols    = (const int*)d_in[7];
    const float* ip_vals  = (const float*)d_in[8];
    const float* user_emb = (const float*)d_in[9];
    const float* item_emb = (const float*)d_in[10];
    const float* Q_user   = (const float*)d_in[11];
    const float* W_user   = (const float*)d_in[12];
    const float* b_user   = (const float*)d_in[13];
    const float* Q_item   = (const float*)d_in[14];
    const float* W_item   = (const float*)d_in[15];
    const float* b_item   = (const float*)d_in[16];
    const float* M_user   = (const float*)d_in[17];
    const float* M_item   = (const float*)d_in[18];
    float* out = (float*)d_out;

    const int NU = 40000, NI = 20000, NN = 60000;
    const int NNZ_ADJ = in_sizes[0], NNZ_UP = in_sizes[3], NNZ_IP = in_sizes[6];
    const size_t ROWF = (size_t)D_DIM;       // floats per row
    const size_t MAT  = (size_t)NN * ROWF;   // 15,360,000 floats

    float* agg  = (float*)d_ws;   // 60000 x 256 (later reused as pair SPMM buf)
    float* embb = agg  + MAT;     // 60000 x 256
    float* hbuf = embb + MAT;     // 60000 x 256

    dim3 blk(256);

    // 1) agg = 0 ; agg = Adj @ pre0
    zero_f32<<<(unsigned)((MAT / 4 + 255) / 256), blk, 0, stream>>>((float4*)agg, (long)(MAT / 4));
    spmm_atomic<<<(NNZ_ADJ + 7) / 8, blk, 0, stream>>>(
        adj_rows, adj_cols, adj_vals, user_emb, item_emb, NU, agg, NNZ_ADJ);

    // 2) emb = tanh(agg @ Q)  (user rows with Q_user, item rows with Q_item)
    gemm_tanh_wmma<<<dim3((NU + 63) / 64, 2), blk, 0, stream>>>(
        agg, nullptr, Q_user, nullptr, nullptr, embb, NU, 0);
    gemm_tanh_wmma<<<dim3((NI + 63) / 64, 2), blk, 0, stream>>>(
        agg + (size_t)NU * ROWF, nullptr, Q_item, nullptr, nullptr,
        embb + (size_t)NU * ROWF, NI, 0);

    // 3) h = tanh([pre0, emb] @ W + b)  -> hbuf
    gemm_tanh_wmma<<<dim3((NU + 63) / 64, 2), blk, 0, stream>>>(
        user_emb, embb, W_user, W_user + (size_t)D_DIM * D_DIM, b_user,
        hbuf, NU, 0);
    gemm_tanh_wmma<<<dim3((NI + 63) / 64, 2), blk, 0, stream>>>(
        item_emb, embb + (size_t)NU * ROWF, W_item,
        W_item + (size_t)D_DIM * D_DIM, b_item,
        hbuf + (size_t)NU * ROWF, NI, 0);

    // 4) out = l2norm(h)   (u_g rows 0..NU, i_g rows NU..NN)
    l2norm_rows<<<(NN + 7) / 8, blk, 0, stream>>>(hbuf, out, NN);

    // 5) pair SPMMs into recycled agg buffer
    zero_f32<<<(unsigned)((MAT / 4 + 255) / 256), blk, 0, stream>>>((float4*)agg, (long)(MAT / 4));
    spmm_atomic<<<(NNZ_UP + 7) / 8, blk, 0, stream>>>(
        up_rows, up_cols, up_vals, user_emb, user_emb, NU, agg, NNZ_UP);
    spmm_atomic<<<(NNZ_IP + 7) / 8, blk, 0, stream>>>(
        ip_rows, ip_cols, ip_vals, item_emb, item_emb, NI,
        agg + (size_t)NU * ROWF, NNZ_IP);

    // 6) out += tanh(P @ M)
    gemm_tanh_wmma<<<dim3((NU + 63) / 64, 2), blk, 0, stream>>>(
        agg, nullptr, M_user, nullptr, nullptr, out, NU, 1);
    gemm_tanh_wmma<<<dim3((NI + 63) / 64, 2), blk, 0, stream>>>(
        agg + (size_t)NU * ROWF, nullptr, M_item, nullptr, nullptr,
        out + (size_t)NU * ROWF, NI, 1);
}